// ClusteringModule_28303834480857
// MI455X (gfx1250) — compile-verified
//
#include <hip/hip_runtime.h>

typedef float v2f __attribute__((ext_vector_type(2)));
typedef float v4f __attribute__((ext_vector_type(4)));
typedef float v8f __attribute__((ext_vector_type(8)));
typedef unsigned int u32x4 __attribute__((ext_vector_type(4)));
typedef int i32x4 __attribute__((ext_vector_type(4)));
typedef int i32x8 __attribute__((ext_vector_type(8)));

#define KDIM   512
#define NCENT  256
#define TILESN 16      // 16 tiles * 16 cols = 256 centroids
#define RPW    16      // rows of z per wave
#define KSLICE 32      // K elements staged per TDM transfer
#define NSLICE (KDIM / KSLICE)
#define LDS_STRIDE 34  // 32 dwords payload + 2 dwords TDM pad (bank de-conflict)

// ---- one-time: cnorm2[j] = ||centroid_j||^2 into workspace ----
__global__ __launch_bounds__(256)
void centroid_norm_kernel(const float* __restrict__ cent,
                          float* __restrict__ cnorm2)
{
    const int j = threadIdx.x;          // 256 threads, one centroid each
    const float* row = cent + (size_t)j * KDIM;
    float s = 0.0f;
#pragma unroll 4
    for (int k = 0; k < KDIM; k += 4) {
        v4f v = *(const v4f*)(row + k);
        s += v.x * v.x + v.y * v.y + v.z * v.z + v.w * v.w;
    }
    cnorm2[j] = s;
}

// Issue one TDM transfer: 256 rows x 32 floats (columns k0..k0+31 of all
// centroids) -> LDS, padded +2 dwords per 32-dword row (LDS stride 34).
__device__ __forceinline__ void tdm_issue(const float* gsrc, unsigned int lds_addr)
{
    const unsigned long long ga = (unsigned long long)(uintptr_t)gsrc;
    u32x4 g0;
    g0.x = 1u;                                           // count=1, user descriptor
    g0.y = lds_addr;                                     // LDS byte offset
    g0.z = (unsigned int)ga;                             // global_addr[31:0]
    g0.w = (unsigned int)((ga >> 32) & 0x1FFFFFFu) | (2u << 30); // addr[56:32] | type=2

    i32x8 g1;
    g1[0] = (2 << 16)     // data_size = 4B
          | (1 << 20)     // pad_enable
          | (4 << 22)     // pad_interval: 32 dwords
          | (1 << 25);    // pad_amount: 2 dwords
    g1[1] = (int)((KDIM & 0xFFFF) << 16);                 // tensor_dim0 lo16 @ [63:48]
    g1[2] = (int)((KDIM >> 16) | ((NCENT & 0xFFFF) << 16)); // dim0 hi16, tensor_dim1 lo16
    g1[3] = (int)((NCENT >> 16) | (KSLICE << 16));        // dim1 hi16, tile_dim0 = 32
    g1[4] = (int)(NCENT);                                 // tile_dim1 = 256, tile_dim2 = 0
    g1[5] = KDIM;                                         // tensor_dim0_stride lo32 = 512
    g1[6] = 0;                                            // stride0 hi16, stride1 lo16
    g1[7] = 0;                                            // stride1 hi32 (unused, 2D)

    const i32x4 gz4 = {0, 0, 0, 0};                       // groups 2/3: 2D tensor
    const i32x8 gz8 = {0, 0, 0, 0, 0, 0, 0, 0};          // extra group (clang-23 form)
    __builtin_amdgcn_tensor_load_to_lds(g0, g1, gz4, gz4, gz8, /*cpol=*/0);
}

__global__ __launch_bounds__(256)
void cluster_student_t_kernel(const float* __restrict__ z,
                              const float* __restrict__ cent,
                              const float* __restrict__ cnorm2,
                              float* __restrict__ s_out,
                              float* __restrict__ c_out,
                              int nRows)
{
    __shared__ float ldsbuf[2][NCENT * LDS_STRIDE];      // 2 x 34 KB double buffer

    const int lane        = threadIdx.x & 31;
    const int waveInBlock = threadIdx.x >> 5;
    const int waveId      = blockIdx.x * (blockDim.x >> 5) + waveInBlock;
    const int rowBase     = waveId * RPW;

    const int half = lane >> 4;     // which 16-lane half
    const int l16  = lane & 15;     // row (A) / col (B) within tile
    const int kk   = 2 * half;      // K sub-offset within a K=4 step

    const float* zrow = z + (size_t)(rowBase + l16) * KDIM;

    // flat LDS pointer -> wave-relative LDS byte offset (aperture truncation)
    const unsigned int lds0 = (unsigned int)(uintptr_t)&ldsbuf[0][0];
    const unsigned int lds1 = (unsigned int)(uintptr_t)&ldsbuf[1][0];

    v8f acc[TILESN];
#pragma unroll
    for (int t = 0; t < TILESN; ++t) acc[t] = v8f{};

    float zn2 = 0.0f;               // partial ||z_row||^2 (this half's K slots)

    // prologue: stage first K slice
    if (threadIdx.x < 32) tdm_issue(cent, lds0);

    for (int kb = 0; kb < NSLICE; ++kb) {
        if (threadIdx.x < 32) __builtin_amdgcn_s_wait_tensorcnt(0);
        __syncthreads();            // slice kb ready; everyone done with kb-1
        if (threadIdx.x < 32 && (kb + 1) < NSLICE)
            tdm_issue(cent + (size_t)(kb + 1) * KSLICE,
                      ((kb + 1) & 1) ? lds1 : lds0);

        const float* lbuf = (kb & 1) ? &ldsbuf[1][0] : &ldsbuf[0][0];
        const int k0 = kb * KSLICE;

        v2f afrag[8];
#pragma unroll
        for (int s = 0; s < 8; ++s) {
            afrag[s] = *(const v2f*)(zrow + k0 + s * 4 + kk);
            zn2 += afrag[s].x * afrag[s].x + afrag[s].y * afrag[s].y;
        }
#pragma unroll
        for (int t = 0; t < TILESN; ++t) {
            const float* lrow = lbuf + (t * 16 + l16) * LDS_STRIDE;
#pragma unroll
            for (int s = 0; s < 8; ++s) {
                v2f bfrag = *(const v2f*)(lrow + s * 4 + kk);   // ds_load_b64
                acc[t] = __builtin_amdgcn_wmma_f32_16x16x4_f32(
                    false, afrag[s], false, bfrag,
                    (short)0, acc[t], false, false);
            }
        }
    }

    // ---- combine K-halves of ||z||^2; fetch precomputed ||c||^2 ----
    zn2 += __shfl_xor(zn2, 16, 32);
    float cn2[TILESN];
#pragma unroll
    for (int t = 0; t < TILESN; ++t) cn2[t] = cnorm2[t * 16 + l16];

    // ---- epilogue: d2 -> s_un = 1/(1+||d||), row-sum, argmax ----
    // C/D layout: VGPR i, lanes 0-15 -> (M=i, N=lane); lanes 16-31 -> (M=i+8, N=lane-16)
    float inv_rs[8];
#pragma unroll
    for (int i = 0; i < 8; ++i) {
        const float zr = __shfl(zn2, i + 8 * half, 32);   // ||z_m||^2, m = i + 8*half
        float rs    = 0.0f;
        float bestS = -1.0f;
        int   bestN = 0;
#pragma unroll
        for (int t = 0; t < TILESN; ++t) {
            float dot = acc[t][i];
            float d2  = fmaxf(zr + cn2[t] - 2.0f * dot, 0.0f);
            float sv  = 1.0f / (1.0f + sqrtf(d2));        // ALPHA=1 student-t
            acc[t][i] = sv;                               // reuse acc as s_un storage
            rs += sv;
            if (sv > bestS) { bestS = sv; bestN = t * 16 + l16; }
        }
#pragma unroll
        for (int m = 1; m < 16; m <<= 1) {                // reduce within 16-lane half
            rs += __shfl_xor(rs, m, 32);
            float oS = __shfl_xor(bestS, m, 32);
            int   oN = __shfl_xor(bestN, m, 32);
            if (oS > bestS || (oS == bestS && oN < bestN)) { bestS = oS; bestN = oN; }
        }
        inv_rs[i] = 1.0f / rs;
        if (l16 == 0) {
            c_out[rowBase + i + 8 * half] = (float)bestN; // argmax as float
        }
    }

    // ---- store normalized s ----
#pragma unroll
    for (int t = 0; t < TILESN; ++t) {
#pragma unroll
        for (int i = 0; i < 8; ++i) {
            const int m = rowBase + i + 8 * half;
            s_out[(size_t)m * NCENT + t * 16 + l16] = acc[t][i] * inv_rs[i];
        }
    }
}

extern "C" void kernel_launch(void* const* d_in, const int* in_sizes, int n_in,
                              void* d_out, int out_size, void* d_ws, size_t ws_size,
                              hipStream_t stream) {
    const float* z    = (const float*)d_in[0];   // [nRows, 512]
    const float* cent = (const float*)d_in[1];   // [256, 512]
    const int nRows   = in_sizes[0] / KDIM;      // 65536

    float* s_out  = (float*)d_out;                       // [nRows, 256]
    float* c_out  = s_out + (size_t)nRows * NCENT;       // [nRows] as float
    float* cnorm2 = (float*)d_ws;                        // [256] scratch

    hipLaunchKernelGGL(centroid_norm_kernel, dim3(1), dim3(NCENT), 0, stream,
                       cent, cnorm2);

    const int waves         = nRows / RPW;       // 4096
    const int wavesPerBlock = 8;                 // 256 threads
    const int blocks        = (waves + wavesPerBlock - 1) / wavesPerBlock;

    hipLaunchKernelGGL(cluster_student_t_kernel,
                       dim3(blocks), dim3(wavesPerBlock * 32), 0, stream,
                       z, cent, cnorm2, s_out, c_out, nRows);
}